// DeepEquilibriumModel_34462817583332
// MI455X (gfx1250) — compile-verified
//
#include <hip/hip_runtime.h>
#include <hip/hip_bf16.h>

typedef __attribute__((ext_vector_type(2))) float v2f;
typedef __attribute__((ext_vector_type(8))) float v8f;
typedef __attribute__((ext_vector_type(4))) int   v4i;

#define NB_ROWS 4096
#define NDIM    2048
#define BETA_C  0.8f
#define LAM_C   1e-4f

// ---------------------------------------------------------------------------
// CDNA5 async global->LDS helpers (ASYNCcnt-tracked, no VGPR round trip).
// ROCm 7.2 clang-22 signature: (v4i* global_src, v4i* lds_dst, imm offset, imm cpol)
// with plain generic pointers (backend resolves address spaces).
// ---------------------------------------------------------------------------
#if __has_builtin(__builtin_amdgcn_global_load_async_to_lds_b128)
#define HAVE_ASYNC_LDS 1
__device__ __forceinline__ void async_copy_b128(const float* g, float* l) {
    __builtin_amdgcn_global_load_async_to_lds_b128((v4i*)g, (v4i*)l, 0, 0);
}
__device__ __forceinline__ void wait_async0() {
#if __has_builtin(__builtin_amdgcn_s_wait_asynccnt)
    __builtin_amdgcn_s_wait_asynccnt(0);
#else
    asm volatile("s_wait_asynccnt 0x0" ::: "memory");
#endif
}
#else
#define HAVE_ASYNC_LDS 0
#endif

// ---------------------------------------------------------------------------
// fp32 GEMM via V_WMMA_F32_16X16X4_F32:  Y = A(MxK) @ W(KxN) + addend
//   addMode==0 : addend = bias[col] (broadcast row-vector)
//   addMode==1 : addend = Cadd[row][col] (full matrix)
// Block tile 128x128, 4 waves (wave32), each wave 64x64 = 4x4 WMMA tiles.
// Double-buffered LDS filled with GLOBAL_LOAD_ASYNC_TO_LDS_B128.
// ---------------------------------------------------------------------------
__global__ __launch_bounds__(128) void gemm_f32_wmma(
    const float* __restrict__ A, const float* __restrict__ W,
    const float* __restrict__ bias, const float* __restrict__ Cadd,
    float* __restrict__ Y, int addMode)
{
    constexpr int K = NDIM;
    constexpr int N = NDIM;

#if HAVE_ASYNC_LDS
    __shared__ float As[2][128][36];   // pad 36: 16B-aligned b128, spread banks
    __shared__ float Bs[2][32][128];
#else
    __shared__ float As[1][128][36];
    __shared__ float Bs[1][32][128];
#endif

    const int t    = threadIdx.x;
    const int lane = t & 31;
    const int wave = t >> 5;
    const int wrB  = (wave >> 1) * 64;     // wave row offset inside block tile
    const int wcB  = (wave & 1)  * 64;     // wave col offset inside block tile
    const int rowBase = blockIdx.y * 128;
    const int colBase = blockIdx.x * 128;

    v8f acc[4][4];
    #pragma unroll
    for (int r = 0; r < 4; ++r)
        #pragma unroll
        for (int c = 0; c < 4; ++c)
            acc[r][c] = v8f{};

    // -------- per-buffer compute: 8 k-steps of 4, 16 WMMAs each ----------
    auto compute_tile = [&](int buf) {
        #pragma unroll
        for (int kk = 0; kk < 32; kk += 4) {
            // lanes 0-15 hold K = ks,ks+1 ; lanes 16-31 hold K = ks+2,ks+3
            const int ks = kk + ((lane >> 4) << 1);
            v2f aF[4], bF[4];
            #pragma unroll
            for (int r = 0; r < 4; ++r) {
                const int r_l = wrB + r * 16 + (lane & 15);
                aF[r].x = As[buf][r_l][ks];
                aF[r].y = As[buf][r_l][ks + 1];
            }
            #pragma unroll
            for (int c = 0; c < 4; ++c) {
                const int c_l = wcB + c * 16 + (lane & 15);
                bF[c].x = Bs[buf][ks][c_l];
                bF[c].y = Bs[buf][ks + 1][c_l];
            }
            #pragma unroll
            for (int r = 0; r < 4; ++r)
                #pragma unroll
                for (int c = 0; c < 4; ++c)
                    acc[r][c] = __builtin_amdgcn_wmma_f32_16x16x4_f32(
                        false, aF[r], false, bF[c],
                        (short)0, acc[r][c], false, false);
        }
    };

#if HAVE_ASYNC_LDS
    // -------- async fill of one LDS buffer (16 x b128 per thread) --------
    auto issue_tile = [&](int buf, int kb) {
        #pragma unroll
        for (int j = 0; j < 8; ++j) {              // A tile 128x32
            int f4  = t + j * 128;
            int row = f4 >> 3;
            int k4  = (f4 & 7) << 2;
            async_copy_b128(&A[(size_t)(rowBase + row) * K + kb + k4],
                            &As[buf][row][k4]);
        }
        #pragma unroll
        for (int j = 0; j < 8; ++j) {              // B tile 32x128
            int f4 = t + j * 128;
            int kr = f4 >> 5;
            int c4 = (f4 & 31) << 2;
            async_copy_b128(&W[(size_t)(kb + kr) * N + colBase + c4],
                            &Bs[buf][kr][c4]);
        }
    };

    // prologue: fill buffer 0
    issue_tile(0, 0);
    wait_async0();
    __syncthreads();

    int cur = 0;
    for (int kb = 0; kb < K; kb += 32) {
        if (kb + 32 < K) issue_tile(cur ^ 1, kb + 32);   // prefetch next tile
        compute_tile(cur);                               // 128 WMMAs
        wait_async0();                                   // next buffer landed
        __syncthreads();                                 // all waves done w/ cur
        cur ^= 1;
    }
#else
    // -------- synchronous fallback (single buffer) --------
    for (int kb = 0; kb < K; kb += 32) {
        #pragma unroll
        for (int j = 0; j < 8; ++j) {
            int f4  = t + j * 128;
            int row = f4 >> 3;
            int k4  = (f4 & 7) << 2;
            float4 v = *reinterpret_cast<const float4*>(
                &A[(size_t)(rowBase + row) * K + kb + k4]);
            *reinterpret_cast<float4*>(&As[0][row][k4]) = v;
        }
        #pragma unroll
        for (int j = 0; j < 8; ++j) {
            int f4 = t + j * 128;
            int kr = f4 >> 5;
            int c4 = (f4 & 31) << 2;
            float4 v = *reinterpret_cast<const float4*>(
                &W[(size_t)(kb + kr) * N + colBase + c4]);
            *reinterpret_cast<float4*>(&Bs[0][kr][c4]) = v;
        }
        __syncthreads();
        compute_tile(0);
        __syncthreads();
    }
#endif

    // ---- epilogue: D layout = VGPR v -> M=v (lanes 0-15) / M=v+8 (lanes 16-31)
    const int rOffLane = (lane >> 4) << 3;   // 0 or 8
    #pragma unroll
    for (int r = 0; r < 4; ++r) {
        #pragma unroll
        for (int c = 0; c < 4; ++c) {
            const int colg = colBase + wcB + c * 16 + (lane & 15);
            #pragma unroll
            for (int v = 0; v < 8; ++v) {
                const int rowg = rowBase + wrB + r * 16 + v + rOffLane;
                const size_t idx = (size_t)rowg * N + colg;
                const float addv = addMode ? Cadd[idx] : bias[colg];
                Y[idx] = acc[r][c][v] + addv;
            }
        }
    }
}

// ---------------------------------------------------------------------------
// Elementwise helpers
// ---------------------------------------------------------------------------
__global__ void zero_kernel(float* __restrict__ p, size_t n)
{
    size_t i = (size_t)blockIdx.x * blockDim.x + threadIdx.x;
    if (i < n) p[i] = 0.0f;
}

// simple step (i < 3): f = tanh(y); g = f - z; store g; z_new = z + beta*g
// (y and g_out may alias: single read then write per element)
__global__ void simple_update(const float* __restrict__ y,
                              const float* __restrict__ z,
                              float* __restrict__ g_out,
                              float* __restrict__ z_new, size_t n)
{
    size_t i = (size_t)blockIdx.x * blockDim.x + threadIdx.x;
    if (i < n) {
        float zi = z[i];
        float g  = tanhf(y[i]) - zi;
        g_out[i] = g;
        z_new[i] = zi + BETA_C * g;
    }
}

// Anderson phase 1: per-row 2x2 Gram + rhs, solve for gamma. Also finalizes g.
__global__ void anderson_dots(const float* __restrict__ y,
                              float* __restrict__ g_out,
                              const float* __restrict__ z,
                              const float* __restrict__ gp1,
                              const float* __restrict__ gp2,
                              float* __restrict__ gamma)
{
    __shared__ float red[5][256];
    const int row = blockIdx.x;
    const int t   = threadIdx.x;
    const size_t base = (size_t)row * NDIM;

    float s00 = 0.f, s01 = 0.f, s11 = 0.f, r0 = 0.f, r1 = 0.f;
    for (int j = t; j < NDIM; j += 256) {
        float g  = tanhf(y[base + j]) - z[base + j];
        g_out[base + j] = g;
        float d0 = g - gp1[base + j];
        float d1 = g - gp2[base + j];
        s00 += d0 * d0; s01 += d0 * d1; s11 += d1 * d1;
        r0  += d0 * g;  r1  += d1 * g;
    }
    red[0][t] = s00; red[1][t] = s01; red[2][t] = s11; red[3][t] = r0; red[4][t] = r1;
    __syncthreads();
    for (int st = 128; st > 0; st >>= 1) {
        if (t < st)
            #pragma unroll
            for (int q = 0; q < 5; ++q) red[q][t] += red[q][t + st];
        __syncthreads();
    }
    if (t == 0) {
        float a  = red[0][0] + LAM_C;
        float b_ = red[1][0];
        float c  = red[1][0];
        float d_ = red[2][0] + LAM_C;
        float R0 = red[3][0], R1 = red[4][0];
        float det = a * d_ - b_ * c + 1e-8f;
        gamma[2 * row + 0] = (d_ * R0 - b_ * R1) / det;
        gamma[2 * row + 1] = (-c * R0 + a * R1) / det;
    }
}

// Anderson phase 2: z_new = z + beta*g - sum_k gamma_k * ((z - zp_k) + beta*(g - gp_k))
__global__ void anderson_update(const float* __restrict__ z,
                                const float* __restrict__ g,
                                const float* __restrict__ zp1,
                                const float* __restrict__ zp2,
                                const float* __restrict__ gp1,
                                const float* __restrict__ gp2,
                                const float* __restrict__ gamma,
                                float* __restrict__ z_new)
{
    const int row = blockIdx.y;
    const int col = blockIdx.x * 256 + threadIdx.x;
    const size_t i = (size_t)row * NDIM + col;
    const float g0 = gamma[2 * row + 0];
    const float g1 = gamma[2 * row + 1];
    const float zi = z[i], gi = g[i];
    float corr = g0 * ((zi - zp1[i]) + BETA_C * (gi - gp1[i]))
               + g1 * ((zi - zp2[i]) + BETA_C * (gi - gp2[i]));
    z_new[i] = zi + BETA_C * gi - corr;
}

// ---------------------------------------------------------------------------
// Host orchestration
// ---------------------------------------------------------------------------
extern "C" void kernel_launch(void* const* d_in, const int* in_sizes, int n_in,
                              void* d_out, int out_size, void* d_ws, size_t ws_size,
                              hipStream_t stream) {
    const float* X    = (const float*)d_in[0];  // 4096 x 2048
    const float* Wz   = (const float*)d_in[1];  // 2048 x 2048
    const float* Wx   = (const float*)d_in[2];  // 2048 x 2048
    const float* bias = (const float*)d_in[3];  // 2048
    float* out = (float*)d_out;
    float* ws  = (float*)d_ws;

    const size_t NB = (size_t)NB_ROWS * NDIM;   // 8,388,608 elements (32 MB)
    float* C0    = ws;
    float* zb[3] = { ws + 1 * NB, ws + 2 * NB, ws + 3 * NB };
    float* gb[3] = { ws + 4 * NB, ws + 5 * NB, ws + 6 * NB };
    float* gamma = ws + 7 * NB;                 // 4096 * 2 floats

    dim3 gemmGrid(NDIM / 128, NB_ROWS / 128);   // (16, 32)
    dim3 gemmBlk(128);
    const int eBlk = 256;
    const int eGrid = (int)(NB / eBlk);         // 32768
    dim3 updGrid(NDIM / 256, NB_ROWS);          // (8, 4096)

    // z0 = 0 (only zcur needs zeroing; histories are fully written before use)
    zero_kernel<<<eGrid, eBlk, 0, stream>>>(zb[0], NB);

    // C0 = X @ Wx + b   (loop-invariant)
    gemm_f32_wmma<<<gemmGrid, gemmBlk, 0, stream>>>(X, Wx, bias, nullptr, C0, 0);

    for (int i = 0; i < 6; ++i) {
        float* zc = zb[i % 3];
        float* gc = gb[i % 3];
        float* zn = (i == 5) ? out : zb[(i + 1) % 3];

        const float* ycur;
        if (i == 0) {
            ycur = C0;                          // z=0 -> y = C0, skip GEMM
        } else {
            // y = z @ Wz + C0, written into gc (rewritten in-place as g below)
            gemm_f32_wmma<<<gemmGrid, gemmBlk, 0, stream>>>(zc, Wz, nullptr, C0, gc, 1);
            ycur = gc;
        }

        if (i < 3) {
            simple_update<<<eGrid, eBlk, 0, stream>>>(ycur, zc, gc, zn, NB);
        } else {
            float* zp1 = zb[(i + 2) % 3];
            float* zp2 = zb[(i + 1) % 3];
            float* gp1 = gb[(i + 2) % 3];
            float* gp2 = gb[(i + 1) % 3];
            anderson_dots<<<NB_ROWS, 256, 0, stream>>>(ycur, gc, zc, gp1, gp2, gamma);
            anderson_update<<<updGrid, 256, 0, stream>>>(zc, gc, zp1, zp2, gp1, gp2,
                                                         gamma, zn);
        }
    }
}